// GAT_56341380989571
// MI455X (gfx1250) — compile-verified
//
#include <hip/hip_runtime.h>
#include <math.h>

// Problem constants (from the reference)
#define N_NODES 100000
#define N_EDGES 800000
#define N_GRAPH 100
#define F_IN    64
#define DIM     128
#define NHEAD   8
#define HD      16
#define NLAYER  3
#define SLOPE   0.2f

typedef float v2f __attribute__((ext_vector_type(2)));
typedef float v8f __attribute__((ext_vector_type(8)));

__device__ __forceinline__ float lrelu(float x) { return x > 0.f ? x : SLOPE * x; }

// ---------------------------------------------------------------------------
// CDNA5 f32 WMMA: D(16x16) = A(16x4) * B(4x16) + C, wave32.
// A layout: lanes 0-15 hold rows M=0..15; VGPR0/1 = K = 2*half + {0,1}.
// B layout: lanes 0-15 hold cols N=0..15; VGPR0/1 = K = 2*half + {0,1}.
// C/D layout: VGPR r = row (r + 8*half), col = lane&15.
// ---------------------------------------------------------------------------
__device__ __forceinline__ v8f wmma_f32_k4(v2f a, v2f b, v8f c) {
#if __has_builtin(__builtin_amdgcn_wmma_f32_16x16x4_f32)
  return __builtin_amdgcn_wmma_f32_16x16x4_f32(false, a, false, b, (short)0, c,
                                               false, false);
#else
  // Shuffle-based emulation with identical register layout (fallback only).
  const int lane = threadIdx.x & 31;
  const int half = lane >> 4;
  const int l16  = lane & 15;
  #pragma unroll
  for (int r = 0; r < 8; ++r) {
    const int m = r + 8 * half;
    float cv = c[r];
    #pragma unroll
    for (int k = 0; k < 4; ++k) {
      const int srcA = (m & 15) | ((k >> 1) << 4);
      const int srcB = l16 | ((k >> 1) << 4);
      const float av = __shfl((k & 1) ? a.y : a.x, srcA, 32);
      const float bv = __shfl((k & 1) ? b.y : b.x, srcB, 32);
      cv = fmaf(av, bv, cv);
    }
    c[r] = cv;
  }
  return c;
#endif
}

// ---------------------------------------------------------------------------
// Tiled GEMM: C[M,128] = act(A[M,K] @ B[K,128] + bias (+ res))
// 256 threads = 8 waves; each wave owns one 16-wide column tile; block owns a
// 16-row tile. A tile staged in LDS (stride K+1 to spread banks).
// ---------------------------------------------------------------------------
template <int K, int ACT /*0 none,1 relu,2 lrelu*/, bool RES>
__global__ __launch_bounds__(256) void gemm_tile(
    const float* __restrict__ A, const float* __restrict__ B,
    const float* __restrict__ bias, const float* __restrict__ res,
    float* __restrict__ C) {
  constexpr int KP = K + 1;
  __shared__ float As[16 * KP];
  const int row0 = blockIdx.x * 16;
  const int tid  = threadIdx.x;

  for (int i = tid; i < 16 * K; i += 256) {
    const int r = i / K, c = i - r * K;
    As[r * KP + c] = A[(size_t)(row0 + r) * K + c];
  }
  __syncthreads();

  const int wave = tid >> 5;
  const int lane = tid & 31;
  const int half = lane >> 4;
  const int l16  = lane & 15;
  const int col0 = wave * 16;

  v8f acc = {0.f, 0.f, 0.f, 0.f, 0.f, 0.f, 0.f, 0.f};
  #pragma unroll
  for (int kk = 0; kk < K; kk += 4) {
    const int ka = kk + 2 * half;
    v2f a, b;
    a.x = As[l16 * KP + ka];
    a.y = As[l16 * KP + ka + 1];
    b.x = B[(size_t)ka * DIM + col0 + l16];
    b.y = B[(size_t)(ka + 1) * DIM + col0 + l16];
    acc = wmma_f32_k4(a, b, acc);
  }

  const int col = col0 + l16;
  const float bv = bias ? bias[col] : 0.f;
  #pragma unroll
  for (int r = 0; r < 8; ++r) {
    const size_t row = (size_t)row0 + r + 8 * half;
    float v = acc[r] + bv;
    if (RES) v += res[row * DIM + col];
    if (ACT == 1) v = fmaxf(v, 0.f);
    else if (ACT == 2) v = lrelu(v);
    C[row * DIM + col] = v;
  }
}

// ---------------------------------------------------------------------------
__global__ void fill_f32(float* __restrict__ p, float v, int n) {
  const int i = blockIdx.x * blockDim.x + threadIdx.x;
  if (i < n) p[i] = v;
}

// al_s[n,h] = <hp[n, h*16 : h*16+16], a_src[h]>, same for al_d
__global__ void attn_coef(const float* __restrict__ hp,
                          const float* __restrict__ a_src,
                          const float* __restrict__ a_dst,
                          float* __restrict__ al_s, float* __restrict__ al_d) {
  const int i = blockIdx.x * blockDim.x + threadIdx.x;
  if (i >= N_NODES * NHEAD) return;
  const int n = i >> 3, h = i & 7;
  const float4* hv = reinterpret_cast<const float4*>(hp + (size_t)n * DIM + h * HD);
  const float4* as = reinterpret_cast<const float4*>(a_src + h * HD);
  const float4* ad = reinterpret_cast<const float4*>(a_dst + h * HD);
  float ss = 0.f, sd = 0.f;
  #pragma unroll
  for (int q = 0; q < 4; ++q) {
    const float4 x = hv[q], u = as[q], w = ad[q];
    ss += x.x * u.x + x.y * u.y + x.z * u.z + x.w * u.w;
    sd += x.x * w.x + x.y * w.y + x.z * w.z + x.w * w.w;
  }
  al_s[i] = ss;
  al_d[i] = sd;
}

__device__ __forceinline__ void atomicMaxF(float* addr, float v) {
  if (v >= 0.f) atomicMax(reinterpret_cast<int*>(addr), __float_as_int(v));
  else atomicMin(reinterpret_cast<unsigned int*>(addr), __float_as_uint(v));
}

__device__ __forceinline__ void atomAddF(float* p, float v) {
  unsafeAtomicAdd(p, v);  // hardware global_atomic_add_f32
}

__device__ __forceinline__ float edge_logit(const float* al_s, const float* al_d,
                                            int s, int r, int h) {
  return lrelu(al_s[s * NHEAD + h] + al_d[r * NHEAD + h]);
}

// Pass 1: per-(receiver, head) max of logits
__global__ void edge_max_k(const int* __restrict__ snd, const int* __restrict__ rcv,
                           const float* __restrict__ al_s, const float* __restrict__ al_d,
                           float* __restrict__ m) {
  const int i = blockIdx.x * blockDim.x + threadIdx.x;
  if (i >= N_EDGES * NHEAD) return;
  const int e = i >> 3, h = i & 7;
  const int s = snd[e], r = rcv[e];
  atomicMaxF(&m[r * NHEAD + h], edge_logit(al_s, al_d, s, r, h));
}

// Pass 2: per-(receiver, head) sum of exp(logit - max)
__global__ void edge_expsum_k(const int* __restrict__ snd, const int* __restrict__ rcv,
                              const float* __restrict__ al_s, const float* __restrict__ al_d,
                              const float* __restrict__ m, float* __restrict__ ssum) {
  const int i = blockIdx.x * blockDim.x + threadIdx.x;
  if (i >= N_EDGES * NHEAD) return;
  const int e = i >> 3, h = i & 7;
  const int s = snd[e], r = rcv[e];
  const float lg = edge_logit(al_s, al_d, s, r, h);
  atomAddF(&ssum[r * NHEAD + h], __expf(lg - m[r * NHEAD + h]));
}

// Pass 3: one wave per edge; lane j handles 4 cols (head = j/4); scatter-add.
__global__ __launch_bounds__(256) void edge_agg_k(
    const int* __restrict__ snd, const int* __restrict__ rcv,
    const float* __restrict__ al_s, const float* __restrict__ al_d,
    const float* __restrict__ m, const float* __restrict__ ssum,
    const float* __restrict__ hp, float* __restrict__ agg) {
  const int gtid = blockIdx.x * blockDim.x + threadIdx.x;
  const int e = gtid >> 5;
  if (e >= N_EDGES) return;
  const int j = threadIdx.x & 31;
  const int s = snd[e], r = rcv[e];
  const int h = j >> 2;
  const float lg = edge_logit(al_s, al_d, s, r, h);
  const float w = __expf(lg - m[r * NHEAD + h]) / (ssum[r * NHEAD + h] + 1e-16f);
  const float4 x = reinterpret_cast<const float4*>(hp + (size_t)s * DIM)[j];
  float* dst = agg + (size_t)r * DIM + 4 * j;
  atomAddF(dst + 0, x.x * w);
  atomAddF(dst + 1, x.y * w);
  atomAddF(dst + 2, x.z * w);
  atomAddF(dst + 3, x.w * w);
}

// Segment-mean pooling honoring the n_node array
__global__ void pool_k(const float* __restrict__ h, const int* __restrict__ n_node,
                       float* __restrict__ pooled) {
  const int g = blockIdx.x, c = threadIdx.x;  // blockDim = DIM
  int start = 0;
  for (int i = 0; i < g; ++i) start += n_node[i];
  const int cnt = n_node[g];
  float s = 0.f;
  for (int i = 0; i < cnt; ++i) s += h[(size_t)(start + i) * DIM + c];
  pooled[g * DIM + c] = s / (float)cnt;
}

__global__ void mlp_dense(const float* __restrict__ in, const float* __restrict__ W,
                          const float* __restrict__ b, float* __restrict__ outp,
                          int rows) {
  const int i = blockIdx.x * blockDim.x + threadIdx.x;
  if (i >= rows * DIM) return;
  const int g = i >> 7, c = i & (DIM - 1);
  float s = b[c];
  for (int k = 0; k < DIM; ++k) s = fmaf(in[g * DIM + k], W[k * DIM + c], s);
  outp[i] = lrelu(s);
}

__global__ void mlp_out(const float* __restrict__ in, const float* __restrict__ W3,
                        const float* __restrict__ b3, float* __restrict__ outp,
                        int rows) {
  const int g = blockIdx.x * blockDim.x + threadIdx.x;
  if (g >= rows) return;
  float s = b3[0];
  for (int k = 0; k < DIM; ++k) s = fmaf(in[g * DIM + k], W3[k], s);
  outp[g] = s;
}

// ---------------------------------------------------------------------------
extern "C" void kernel_launch(void* const* d_in, const int* in_sizes, int n_in,
                              void* d_out, int out_size, void* d_ws, size_t ws_size,
                              hipStream_t stream) {
  (void)in_sizes; (void)n_in; (void)out_size; (void)ws_size;

  const float* x       = (const float*)d_in[0];
  const int*   senders = (const int*)d_in[1];
  const int*   recv    = (const int*)d_in[2];
  const int*   n_node  = (const int*)d_in[3];
  const float* W_in    = (const float*)d_in[4];
  const float* b_in    = (const float*)d_in[5];
  const float* W_gat   = (const float*)d_in[6];
  const float* a_src   = (const float*)d_in[7];
  const float* a_dst   = (const float*)d_in[8];
  const float* W_skip  = (const float*)d_in[9];
  const float* b_skip  = (const float*)d_in[10];
  const float* W1      = (const float*)d_in[11];
  const float* b1      = (const float*)d_in[12];
  const float* W2      = (const float*)d_in[13];
  const float* b2      = (const float*)d_in[14];
  const float* W3      = (const float*)d_in[15];
  const float* b3      = (const float*)d_in[16];
  float* out = (float*)d_out;

  float* ws = (float*)d_ws;
  float* h      = ws; ws += (size_t)N_NODES * DIM;
  float* hp     = ws; ws += (size_t)N_NODES * DIM;
  float* agg    = ws; ws += (size_t)N_NODES * DIM;
  float* al_s   = ws; ws += (size_t)N_NODES * NHEAD;
  float* al_d   = ws; ws += (size_t)N_NODES * NHEAD;
  float* mbuf   = ws; ws += (size_t)N_NODES * NHEAD;
  float* sbuf   = ws; ws += (size_t)N_NODES * NHEAD;
  float* pooled = ws; ws += N_GRAPH * DIM;
  float* z1     = ws; ws += N_GRAPH * DIM;
  float* z2     = ws; ws += N_GRAPH * DIM;

  const int rowTiles = N_NODES / 16;                       // 6250 (N % 16 == 0)
  const int nhThreads = N_NODES * NHEAD;
  const int ehThreads = N_EDGES * NHEAD;

  // h = relu(x @ W_in + b_in)
  gemm_tile<F_IN, 1, false><<<rowTiles, 256, 0, stream>>>(x, W_in, b_in, nullptr, h);

  for (int l = 0; l < NLAYER; ++l) {
    const float* Wg = W_gat + (size_t)l * DIM * DIM;
    const float* Ws = W_skip + (size_t)l * DIM * DIM;
    const float* bs = b_skip + (size_t)l * DIM;
    const float* as = a_src + (size_t)l * NHEAD * HD;
    const float* ad = a_dst + (size_t)l * NHEAD * HD;

    // hp = h @ W_gat[l]
    gemm_tile<DIM, 0, false><<<rowTiles, 256, 0, stream>>>(h, Wg, nullptr, nullptr, hp);

    attn_coef<<<(nhThreads + 255) / 256, 256, 0, stream>>>(hp, as, ad, al_s, al_d);

    fill_f32<<<(nhThreads + 255) / 256, 256, 0, stream>>>(mbuf, -INFINITY, nhThreads);
    fill_f32<<<(nhThreads + 255) / 256, 256, 0, stream>>>(sbuf, 0.f, nhThreads);
    fill_f32<<<((N_NODES * DIM) + 255) / 256, 256, 0, stream>>>(agg, 0.f, N_NODES * DIM);

    edge_max_k<<<(ehThreads + 255) / 256, 256, 0, stream>>>(senders, recv, al_s, al_d, mbuf);
    edge_expsum_k<<<(ehThreads + 255) / 256, 256, 0, stream>>>(senders, recv, al_s, al_d, mbuf, sbuf);
    edge_agg_k<<<(N_EDGES * 32) / 256, 256, 0, stream>>>(senders, recv, al_s, al_d,
                                                         mbuf, sbuf, hp, agg);

    // h = lrelu(h + agg @ W_skip[l] + b_skip[l])   (in-place residual)
    gemm_tile<DIM, 2, true><<<rowTiles, 256, 0, stream>>>(agg, Ws, bs, h, h);
  }

  pool_k<<<N_GRAPH, DIM, 0, stream>>>(h, n_node, pooled);
  mlp_dense<<<(N_GRAPH * DIM + 255) / 256, 256, 0, stream>>>(pooled, W1, b1, z1, N_GRAPH);
  mlp_dense<<<(N_GRAPH * DIM + 255) / 256, 256, 0, stream>>>(z1, W2, b2, z2, N_GRAPH);
  mlp_out<<<1, 128, 0, stream>>>(z2, W3, b3, out, N_GRAPH);
}